// Sponge_58488864637517
// MI455X (gfx1250) — compile-verified
//
#include <hip/hip_runtime.h>

// ---------------------------------------------------------------------------
// Sponge network for MI455X (gfx1250, wave32).
//   Kernel A: precompute {cos,sin} table for angles (48x256) into d_ws once
//             per launch -> removes ~200M redundant per-row sincos evals.
//   Kernel B: one workgroup (512 threads = 16 wave32) per batch row.
//             mem (5120 f32) + double-buffered sponge (2x512 f32) in LDS
//             (24 KB of 320 KB/WGP). X row staged via the CDNA5 async
//             global->LDS path (global_load_async_to_lds_b128 + ASYNCcnt).
// HBM traffic: 32 MB in + 8 MB out  ->  ~1.7 us floor @ 23.3 TB/s.
// ---------------------------------------------------------------------------

#ifndef __has_builtin
#define __has_builtin(x) 0
#endif

#if __has_builtin(__builtin_amdgcn_global_load_async_to_lds_b128)
#define SPONGE_ASYNC_LDS 1
#else
#define SPONGE_ASYNC_LDS 0
#endif

namespace sponge {
constexpr int INPUT   = 1024;
constexpr int OUT     = 256;
constexpr int SP      = 512;
constexpr int ACT     = 64;
constexpr int REC     = 32;
constexpr int NDEPTH  = 48;
constexpr int STORE   = ACT + REC;               // 96
constexpr int START   = INPUT - SP;              // 512
constexpr int MEMSZ   = START + NDEPTH * STORE;  // 5120
constexpr int EXCH    = SP / 2;                  // 256
constexpr int NANG    = NDEPTH * EXCH;           // 12288
} // namespace sponge

#if SPONGE_ASYNC_LDS
typedef int v4i __attribute__((ext_vector_type(4)));
typedef v4i __attribute__((address_space(1)))* gv4i_ptr;  // global int4*
typedef v4i __attribute__((address_space(3)))* lv4i_ptr;  // LDS int4*

__device__ __forceinline__ gv4i_ptr to_global_v4(const void* p) {
  // flat and global addresses are bit-identical for global memory.
  return (gv4i_ptr)(uintptr_t)p;
}
__device__ __forceinline__ lv4i_ptr to_lds_v4(void* p) {
  // generic -> LDS: ptrtoint/inttoptr round-trip; InferAddressSpaces folds
  // this back to the original addrspace(3) pointer of the __shared__ array.
  return (lv4i_ptr)(uintptr_t)p;
}
#endif

// ---- Kernel A: cos/sin table ----------------------------------------------
__global__ __launch_bounds__(256) void cs_init_kernel(
    const float* __restrict__ angles, float2* __restrict__ cstab, int n) {
  int idx = blockIdx.x * blockDim.x + threadIdx.x;
  if (idx < n) {
    float s, c;
    sincosf(angles[idx], &s, &c);
    cstab[idx] = make_float2(c, s);
  }
}

// ---- Kernel B: sponge -----------------------------------------------------
template <bool USE_TAB>
__global__ __launch_bounds__(512) void sponge_kernel(
    const float*  __restrict__ X,          // [B,1024]
    const float*  __restrict__ scales,     // [1024]
    const float*  __restrict__ angles,     // [48,256]
    const float2* __restrict__ cstab,      // [48*256] {cos,sin} (if USE_TAB)
    const float*  __restrict__ bias,       // [48,64]
    const int*    __restrict__ recall_idx, // [48,32]
    const int*    __restrict__ output_idx, // [256]
    float* __restrict__ out)               // [B,256]
{
  using namespace sponge;

  __shared__ float mem[MEMSZ]; // 20 KB
  __shared__ float spA[SP];    // current sponge
  __shared__ float spB[SP];    // post-rotation+shuffle "sp"

  const int t   = threadIdx.x;          // 0..511
  const int row = blockIdx.x;
  const float* __restrict__ Xrow = X + (size_t)row * INPUT;

  // ---- stage X row into LDS and apply scales -----------------------------
#if SPONGE_ASYNC_LDS
  if (t < 256) {
    // 256 lanes x 16B = 4 KB = one full row. Lo half -> spB, hi half -> mem.
    const float* g;
    float*       l;
    if (t < 128) { g = Xrow + 4 * t;               l = spB + 4 * t;         }
    else         { g = Xrow + SP + 4 * (t - 128);  l = mem + 4 * (t - 128); }
    __builtin_amdgcn_global_load_async_to_lds_b128(to_global_v4(g),
                                                   to_lds_v4(l),
                                                   /*offset=*/0, /*cpol=*/0);
  }
#if __has_builtin(__builtin_amdgcn_s_wait_asynccnt)
  __builtin_amdgcn_s_wait_asynccnt(0);
#else
  asm volatile("s_wait_asynccnt 0" ::: "memory");
#endif
  __syncthreads();
  spA[t] = spB[t] * scales[t];
  mem[t] = mem[t] * scales[SP + t];
#else
  spA[t] = Xrow[t]      * scales[t];
  mem[t] = Xrow[SP + t] * scales[SP + t];
#endif
  __syncthreads();

  // Per-thread loop-invariant shuffle indices:
  //   sp[t] = rot[k],  k = SHUFFLE[t] = 2t (t<256) else 2t-511.
  const int  k     = (t < EXCH) ? (t << 1) : ((t << 1) - (SP - 1));
  const bool upper = (k >= EXCH);
  const int  ka    = upper ? (k - EXCH) : k;   // angle index
  const int  kp    = upper ? (k - EXCH) : (k + EXCH);  // partner element

  // ---- 48 rounds ---------------------------------------------------------
  for (int i = 0; i < NDEPTH; ++i) {
    // Pass 1: fused rotate + riffle-shuffle.
    float c, s;
    if (USE_TAB) {
      const float2 cs = cstab[i * EXCH + ka];  // L2-resident 96 KB table
      c = cs.x; s = cs.y;
    } else {
      sincosf(angles[i * EXCH + ka], &s, &c);
    }
    const float v = upper ? fmaf(c, spA[k], -(s * spA[kp]))   // -s*x1 + c*x2
                          : fmaf(c, spA[k],   s * spA[kp]);   //  c*x1 + s*x2
    spB[t] = v;
    if (t < STORE) mem[START + i * STORE + t] = v;  // mem slice = sp[0:96]
    __syncthreads();  // pass-1 reads of spA done; spB + mem slice visible

    // Pass 2: assemble next sponge =
    //   [ sp[96:128] | sp[192:512] | relu(act) | relu(-act) | recall ]
    float nv;
    if (t < 32) {
      nv = spB[96 + t];
    } else if (t < 352) {
      nv = spB[t + 160];                       // sp[192 + (t-32)]
    } else if (t < 416) {
      float a = spB[t - 224] + bias[i * ACT + (t - 352)]; // sp[128+..]
      nv = fmaxf(a, 0.0f);
    } else if (t < 480) {
      float a = spB[t - 288] + bias[i * ACT + (t - 416)];
      nv = fmaxf(-a, 0.0f);
    } else {
      nv = mem[recall_idx[i * REC + (t - 480)]];
    }
    spA[t] = nv;
    __syncthreads();  // next round's pass-1 reads spA
  }

  // ---- output gather -----------------------------------------------------
  if (t < OUT) {
    out[(size_t)row * OUT + t] = mem[output_idx[t]];
  }
}

extern "C" void kernel_launch(void* const* d_in, const int* in_sizes, int n_in,
                              void* d_out, int out_size, void* d_ws,
                              size_t ws_size, hipStream_t stream) {
  using namespace sponge;
  const float* X          = (const float*)d_in[0];
  const float* scales     = (const float*)d_in[1];
  const float* angles     = (const float*)d_in[2];
  const float* bias       = (const float*)d_in[3];
  const int*   recall_idx = (const int*)d_in[4];
  const int*   output_idx = (const int*)d_in[5];
  float* out = (float*)d_out;

  const int rows = in_sizes[0] / INPUT;  // 8192
  const size_t tab_bytes = (size_t)NANG * sizeof(float2);  // 96 KB

  if (ws_size >= tab_bytes && d_ws != nullptr) {
    float2* cstab = (float2*)d_ws;
    cs_init_kernel<<<(NANG + 255) / 256, 256, 0, stream>>>(angles, cstab, NANG);
    sponge_kernel<true><<<rows, 512, 0, stream>>>(
        X, scales, angles, cstab, bias, recall_idx, output_idx, out);
  } else {
    sponge_kernel<false><<<rows, 512, 0, stream>>>(
        X, scales, angles, nullptr, bias, recall_idx, output_idx, out);
  }
}